// Graph_Layer_bidirection_36507222016271
// MI455X (gfx1250) — compile-verified
//
#include <hip/hip_runtime.h>

typedef __attribute__((ext_vector_type(2))) float v2f;
typedef __attribute__((ext_vector_type(8))) float v8f;

#define D 128
#define D2 (D * D)

__device__ __forceinline__ v2f ld2(const float* __restrict__ p) {
    return *(const v2f* __restrict__)p;
}

// ---------------------------------------------------------------------------
// Fold the 3D->D message weights through the node/edge projections:
//   mat 0: A_s2t = Wm_s2t[:,   0:128] @ Wn_s2t
//   mat 1: B_s2t = Wm_s2t[:, 128:256] @ Wn_s2t
//   mat 2: C_s2t = Wm_s2t[:, 256:384] @ We_s2t
//   mat 3..5: same with t2s params
// Wf[mat][k][c] row-major, 6 * 128 * 128 floats.
// ---------------------------------------------------------------------------
__global__ __launch_bounds__(256) void fuse_weights(
    const float* __restrict__ Wn_s2t, const float* __restrict__ We_s2t,
    const float* __restrict__ Wm_s2t,
    const float* __restrict__ Wn_t2s, const float* __restrict__ We_t2s,
    const float* __restrict__ Wm_t2s,
    float* __restrict__ Wf) {
    int idx = blockIdx.x * blockDim.x + threadIdx.x;   // 6*16384 threads
    if (idx >= 6 * D2) return;
    int mat = idx >> 14;
    int k   = (idx >> 7) & (D - 1);
    int c   = idx & (D - 1);
    const float* Wm = (mat < 3) ? Wm_s2t : Wm_t2s;
    const float* R  = ((mat % 3) == 2) ? ((mat < 3) ? We_s2t : We_t2s)
                                       : ((mat < 3) ? Wn_s2t : Wn_t2s);
    int off = (mat % 3) * D;
    float s = 0.f;
#pragma unroll 8
    for (int m = 0; m < D; ++m)
        s = __builtin_fmaf(Wm[k * (3 * D) + off + m], R[m * D + c], s);
    Wf[idx] = s;
}

// ---------------------------------------------------------------------------
// Edge pass: one wave (32 lanes x float4) per edge.
//   Sx_src[src] += x[dst];  Se_src[src] += ea;  deg_src[src] += 1   (t2s)
//   Sx_dst[dst] += x[src];  Se_dst[dst] += ea;  deg_dst[dst] += 1   (s2t)
// Also streams edge_attr into the second output (tuple copy).
// ---------------------------------------------------------------------------
__global__ __launch_bounds__(256) void edge_scatter(
    const float* __restrict__ x, const int* __restrict__ edge_index,
    const float* __restrict__ edge_attr,
    float* __restrict__ Sx_src, float* __restrict__ Sx_dst,
    float* __restrict__ Se_src, float* __restrict__ Se_dst,
    float* __restrict__ deg_src, float* __restrict__ deg_dst,
    float* __restrict__ edge_out, int nE) {
    const int lane = threadIdx.x & 31;
    const int e = blockIdx.x * (blockDim.x >> 5) + (threadIdx.x >> 5);
    if (e >= nE) return;
    const int src = edge_index[e];
    const int dst = edge_index[nE + e];
    const int c4 = lane * 4;
    const size_t eoff = (size_t)e * D + c4;
    const float4 ea = *(const float4*)(edge_attr + eoff);
    *(float4*)(edge_out + eoff) = ea;
    const float4 xs = *(const float4*)(x + (size_t)src * D + c4);
    const float4 xd = *(const float4*)(x + (size_t)dst * D + c4);
    float* p;
    p = Sx_src + (size_t)src * D + c4;
    atomicAdd(p + 0, xd.x); atomicAdd(p + 1, xd.y);
    atomicAdd(p + 2, xd.z); atomicAdd(p + 3, xd.w);
    p = Sx_dst + (size_t)dst * D + c4;
    atomicAdd(p + 0, xs.x); atomicAdd(p + 1, xs.y);
    atomicAdd(p + 2, xs.z); atomicAdd(p + 3, xs.w);
    p = Se_src + (size_t)src * D + c4;
    atomicAdd(p + 0, ea.x); atomicAdd(p + 1, ea.y);
    atomicAdd(p + 2, ea.z); atomicAdd(p + 3, ea.w);
    p = Se_dst + (size_t)dst * D + c4;
    atomicAdd(p + 0, ea.x); atomicAdd(p + 1, ea.y);
    atomicAdd(p + 2, ea.z); atomicAdd(p + 3, ea.w);
    if (lane == 0) {
        atomicAdd(deg_src + src, 1.0f);
        atomicAdd(deg_dst + dst, 1.0f);
    }
}

// ---------------------------------------------------------------------------
// Node pass on WMMA: one wave owns a 16(node) x 16(feature) tile.
// D = S @ W^T  ==  WMMA(A = S tile, B = W^T tile).
// f32 A layout (ISA 7.12.2): lanes 0-15 rows M, v0=K+0/v1=K+1;
//                            lanes 16-31 rows M, v0=K+2/v1=K+3.
// B = W^T shares the same per-lane addressing with rows of W indexed by N.
// C/D layout: VGPR p, lane l  ->  M = p + 8*(l>>4), N = l&15.
// ---------------------------------------------------------------------------
__global__ __launch_bounds__(256) void node_gemm_wmma(
    const float* __restrict__ x,
    const float* __restrict__ Sx_src, const float* __restrict__ Sx_dst,
    const float* __restrict__ Se_src, const float* __restrict__ Se_dst,
    const float* __restrict__ deg_src, const float* __restrict__ deg_dst,
    const float* __restrict__ Wf,
    const float* __restrict__ bm_s2t, const float* __restrict__ bm_t2s,
    float* __restrict__ out, int nNodes) {
    const int lane = threadIdx.x & 31;
    const int wave = threadIdx.x >> 5;       // 8 waves -> 8 feature tiles
    const int m0 = blockIdx.x * 16;
    if (m0 >= nNodes) return;
    const int n0 = wave * 16;
    const int r = lane & 15;
    const int half = lane >> 4;

    const float* A_s2t = Wf + 0 * D2;
    const float* B_s2t = Wf + 1 * D2;
    const float* C_s2t = Wf + 2 * D2;
    const float* A_t2s = Wf + 3 * D2;
    const float* B_t2s = Wf + 4 * D2;
    const float* C_t2s = Wf + 5 * D2;

    int mrow = m0 + r;
    if (mrow >= nNodes) mrow = nNodes - 1;   // N % 16 == 0 here; safety clamp
    const int rowA = mrow * D;
    const int rowB = (n0 + r) * D;

    v8f acc0 = {};   // Sx_dst@A_s2t^T + Se_dst@C_s2t^T + Sx_src@A_t2s^T + Se_src@C_t2s^T
    v8f acc1 = {};   // x@B_s2t^T  (scaled by deg_dst later)
    v8f acc2 = {};   // x@B_t2s^T  (scaled by deg_src later)

#pragma unroll 4
    for (int k = 0; k < D; k += 4) {
        const int ka = k + 2 * half;
        const v2f aXd = ld2(Sx_dst + rowA + ka);
        const v2f aEd = ld2(Se_dst + rowA + ka);
        const v2f aXs = ld2(Sx_src + rowA + ka);
        const v2f aEs = ld2(Se_src + rowA + ka);
        const v2f aX  = ld2(x      + rowA + ka);
        const v2f bAs = ld2(A_s2t + rowB + ka);
        const v2f bCs = ld2(C_s2t + rowB + ka);
        const v2f bAt = ld2(A_t2s + rowB + ka);
        const v2f bCt = ld2(C_t2s + rowB + ka);
        const v2f bBs = ld2(B_s2t + rowB + ka);
        const v2f bBt = ld2(B_t2s + rowB + ka);
        acc0 = __builtin_amdgcn_wmma_f32_16x16x4_f32(false, aXd, false, bAs, (short)0, acc0, false, false);
        acc0 = __builtin_amdgcn_wmma_f32_16x16x4_f32(false, aEd, false, bCs, (short)0, acc0, false, false);
        acc0 = __builtin_amdgcn_wmma_f32_16x16x4_f32(false, aXs, false, bAt, (short)0, acc0, false, false);
        acc0 = __builtin_amdgcn_wmma_f32_16x16x4_f32(false, aEs, false, bCt, (short)0, acc0, false, false);
        acc1 = __builtin_amdgcn_wmma_f32_16x16x4_f32(false, aX,  false, bBs, (short)0, acc1, false, false);
        acc2 = __builtin_amdgcn_wmma_f32_16x16x4_f32(false, aX,  false, bBt, (short)0, acc2, false, false);
    }

    const int n = n0 + r;
    const float bs = bm_s2t[n];
    const float bt = bm_t2s[n];
#pragma unroll
    for (int p = 0; p < 8; ++p) {
        const int m = m0 + p + 8 * half;
        if (m >= nNodes) continue;
        const float dd = deg_dst[m];
        const float ds = deg_src[m];
        const float val = acc0[p] + dd * (acc1[p] + bs) + ds * (acc2[p] + bt)
                        + 2.0f * x[m * D + n];
        out[m * D + n] = 0.5f * val;
    }
}

extern "C" void kernel_launch(void* const* d_in, const int* in_sizes, int n_in,
                              void* d_out, int out_size, void* d_ws, size_t ws_size,
                              hipStream_t stream) {
    const float* x       = (const float*)d_in[0];
    const int*   ei      = (const int*)d_in[1];
    const float* ea      = (const float*)d_in[2];
    // d_in[3]=memory, d_in[4]=batch_id: unused
    const float* Wn_s2t  = (const float*)d_in[5];
    const float* We_s2t  = (const float*)d_in[6];
    const float* Wm_s2t  = (const float*)d_in[7];
    const float* bm_s2t  = (const float*)d_in[8];
    const float* Wn_t2s  = (const float*)d_in[9];
    const float* We_t2s  = (const float*)d_in[10];
    const float* Wm_t2s  = (const float*)d_in[11];
    const float* bm_t2s  = (const float*)d_in[12];

    const int nNodes = in_sizes[0] / D;
    const int nE     = in_sizes[1] / 2;

    float* node_out = (float*)d_out;              // (N, D)
    float* edge_out = node_out + (size_t)nNodes * D;  // (E, D) tuple copy

    // Workspace layout (floats)
    float* ws      = (float*)d_ws;
    float* Sx_src  = ws;
    float* Sx_dst  = Sx_src + (size_t)nNodes * D;
    float* Se_src  = Sx_dst + (size_t)nNodes * D;
    float* Se_dst  = Se_src + (size_t)nNodes * D;
    float* deg_src = Se_dst + (size_t)nNodes * D;
    float* deg_dst = deg_src + nNodes;
    float* Wf      = deg_dst + nNodes;            // 6 * 128 * 128

    // Zero accumulators + degree counters every call (deterministic).
    hipMemsetAsync(Sx_src, 0,
                   ((size_t)4 * nNodes * D + 2 * (size_t)nNodes) * sizeof(float),
                   stream);

    // 1) Fold weights: 6 x 128x128 composed matrices.
    fuse_weights<<<(6 * D2 + 255) / 256, 256, 0, stream>>>(
        Wn_s2t, We_s2t, Wm_s2t, Wn_t2s, We_t2s, Wm_t2s, Wf);

    // 2) Edge scatter: one wave per edge, 8 edges per 256-thread block.
    const int edgesPerBlock = 256 / 32;
    edge_scatter<<<(nE + edgesPerBlock - 1) / edgesPerBlock, 256, 0, stream>>>(
        x, ei, ea, Sx_src, Sx_dst, Se_src, Se_dst, deg_src, deg_dst,
        edge_out, nE);

    // 3) Node GEMMs on WMMA: one block per 16 nodes, 8 waves cover 128 features.
    node_gemm_wmma<<<(nNodes + 15) / 16, 256, 0, stream>>>(
        x, Sx_src, Sx_dst, Se_src, Se_dst, deg_src, deg_dst, Wf,
        bm_s2t, bm_t2s, node_out, nNodes);
}